// AnatomicalConsistencyLoss_84378927497558
// MI455X (gfx1250) — compile-verified
//
#include <hip/hip_runtime.h>
#include <hip/hip_bf16.h>

// Volume / tiling geometry
#define Wd 160
#define Hd 160
#define Dd 160
#define TB 2                    // batch
#define TX 16
#define TY 16
#define ZC 20                   // z voxels per block
#define TILES_X (Wd / TX)       // 10
#define TILES_Y (Hd / TY)       // 10
#define ZCHUNKS (Dd / ZC)       // 8
#define NBLOCKS (TB * ZCHUNKS * TILES_Y * TILES_X)  // 1600
#define PITCH 20                // LDS row pitch (floats), 18 used + pad
#define PLANE (18 * PITCH)      // one halo plane slot (floats)

typedef __attribute__((ext_vector_type(2))) float v2f;
typedef __attribute__((ext_vector_type(8))) float v8f;

// Generic (flat) LDS address: [63:32]=shared aperture, [31:0]=LDS byte offset.
// GLOBAL_LOAD_ASYNC_TO_LDS adds LDS_BASE itself, so the low 32 bits are exactly
// the VDST operand it wants.
__device__ __forceinline__ unsigned lds_addr(const void* p) {
  return (unsigned)(unsigned long long)p;
}

// Async global->LDS dword copy (GVS mode: saddr base + 32-bit vgpr byte offset).
// Tracked on ASYNCcnt, overlapped with compute; per-lane EXEC masking gives us
// free boundary handling (inactive lanes ds_store zeros instead).
__device__ __forceinline__ void async_load_dword(unsigned lds_off, const float* base,
                                                 unsigned byte_off) {
  asm volatile("global_load_async_to_lds_b32 %0, %1, %2"
               :: "v"(lds_off), "v"(byte_off), "s"(base)
               : "memory");
}

__device__ __forceinline__ void wait_async0() {
  asm volatile("s_wait_asynccnt 0" ::: "memory");
}

__global__ void __launch_bounds__(256)
sobel_loss_tiles(const float* __restrict__ pred,
                 const float* __restrict__ targ,
                 float* __restrict__ partials) {
  // [tensor 0/1][ring slot 0..3][18x18 plane @ pitch 20]
  __shared__ float smem[2 * 4 * PLANE];
  __shared__ float redm[256];
  __shared__ float redc[256];

  const int tid = threadIdx.x;
  const int tx = tid & (TX - 1);
  const int ty = tid >> 4;

  int bid = blockIdx.x;
  const int tix = bid % TILES_X; bid /= TILES_X;
  const int tiy = bid % TILES_Y; bid /= TILES_Y;
  const int zc  = bid % ZCHUNKS; bid /= ZCHUNKS;
  const int bb  = bid;  // batch

  const int x0 = tix * TX;
  const int y0 = tiy * TY;
  const int z0 = zc * ZC;

  const size_t batch_off = (size_t)bb * ((size_t)Dd * Hd * Wd);
  const float* pbase = pred + batch_off;
  const float* tbase = targ + batch_off;

  // Stage halo plane z (z in [-1, Dd]) into ring slot (z+1)&3 for both tensors.
  // In-bounds cells: async global->LDS. Out-of-bounds (conv zero padding): ds zero.
  auto stage = [&](int z) {
    const int slot = (z + 1) & 3;
    float* sp = &smem[slot * PLANE];
    float* st = &smem[(4 + slot) * PLANE];
    const bool zin = (unsigned)z < (unsigned)Dd;
    for (int i = tid; i < 18 * 18; i += 256) {
      const int lx = i % 18;
      const int ly = i / 18;
      const int gx = x0 - 1 + lx;
      const int gy = y0 - 1 + ly;
      const int l  = ly * PITCH + lx;
      if (zin && (unsigned)gx < (unsigned)Wd && (unsigned)gy < (unsigned)Hd) {
        const unsigned goff = (unsigned)(((z * Hd + gy) * Wd + gx) * 4);
        async_load_dword(lds_addr(&sp[l]), pbase, goff);
        async_load_dword(lds_addr(&st[l]), tbase, goff);
      } else {
        sp[l] = 0.0f;
        st[l] = 0.0f;
      }
    }
  };

  // Separable in-plane partials at this thread's (x,y):
  //   A = smooth_y(deriv_x(v)), B = deriv_y(smooth_x(v)), C = smooth_y(smooth_x(v))
  // 9 LDS reads per tensor per plane; z-combination later from a register ring.
  auto plane_abc = [&](int z, int t, float& A, float& B, float& C) {
    const int slot = (z + 1) & 3;
    const float* p = &smem[(t * 4 + slot) * PLANE + ty * PITCH + tx];
    const float a0 = p[0],         a1 = p[1],             a2 = p[2];
    const float b0 = p[PITCH],     b1 = p[PITCH + 1],     b2 = p[PITCH + 2];
    const float c0 = p[2 * PITCH], c1 = p[2 * PITCH + 1], c2 = p[2 * PITCH + 2];
    const float rsa = a0 + 2.f * a1 + a2, rda = a2 - a0;
    const float rsb = b0 + 2.f * b1 + b2, rdb = b2 - b0;
    const float rsc = c0 + 2.f * c1 + c2, rdc = c2 - c0;
    A = rda + 2.f * rdb + rdc;   // smooth over y of deriv_x rows
    B = rsc - rsa;               // deriv over y of smooth_x rows
    C = rsa + 2.f * rsb + rsc;   // smooth over y of smooth_x rows
  };

  stage(z0 - 1);
  stage(z0);
  stage(z0 + 1);
  wait_async0();
  __syncthreads();

  float pA0, pB0, pC0, pA1, pB1, pC1;
  float tA0, tB0, tC0, tA1, tB1, tC1;
  plane_abc(z0 - 1, 0, pA0, pB0, pC0);
  plane_abc(z0,     0, pA1, pB1, pC1);
  plane_abc(z0 - 1, 1, tA0, tB0, tC0);
  plane_abc(z0,     1, tA1, tB1, tC1);

  float magsum = 0.f;
  float cossum = 0.f;

  for (int z = z0; z < z0 + ZC; ++z) {
    // Prefetch plane z+2 asynchronously; it overwrites the slot of plane z-2,
    // whose last readers finished at least one barrier ago.
    if (z + 2 <= z0 + ZC) stage(z + 2);

    float pA2, pB2, pC2, tA2, tB2, tC2;
    plane_abc(z + 1, 0, pA2, pB2, pC2);
    plane_abc(z + 1, 1, tA2, tB2, tC2);

    // z-combination (smooth = [1,2,1], deriv = [-1,0,1])
    const float gxp = pA0 + 2.f * pA1 + pA2;
    const float gyp = pB0 + 2.f * pB1 + pB2;
    const float gzp = pC2 - pC0;
    const float gxt = tA0 + 2.f * tA1 + tA2;
    const float gyt = tB0 + 2.f * tB1 + tB2;
    const float gzt = tC2 - tC0;

    const float sp2 = gxp * gxp + gyp * gyp + gzp * gzp;
    const float st2 = gxt * gxt + gyt * gyt + gzt * gzt;

    // magnitude loss term: (sqrt(|g|^2 + 1e-8) diff)^2
    const float pm = sqrtf(sp2 + 1e-8f);
    const float tm = sqrtf(st2 + 1e-8f);
    const float dm = pm - tm;
    magsum += dm * dm;

    // direction term, faithful to normalize(eps=1e-12) + cosine(eps=1e-8)
    const float npn = sqrtf(sp2);
    const float ntn = sqrtf(st2);
    const float rp = 1.0f / fmaxf(npn, 1e-12f);
    const float rt = 1.0f / fmaxf(ntn, 1e-12f);
    const float dot = (gxp * gxt + gyp * gyt + gzp * gzt) * rp * rt;
    const float na = fmaxf(npn * rp, 1e-8f);
    const float nb = fmaxf(ntn * rt, 1e-8f);
    cossum += dot / (na * nb);

    // rotate the register ring
    pA0 = pA1; pB0 = pB1; pC0 = pC1;
    pA1 = pA2; pB1 = pB2; pC1 = pC2;
    tA0 = tA1; tB0 = tB1; tC0 = tC1;
    tA1 = tA2; tB1 = tB2; tC1 = tC2;

    wait_async0();    // this wave's prefetch landed in LDS
    __syncthreads();  // all waves done -> slot reuse + cross-wave visibility
  }

  // Deterministic block reduction -> per-block partials (no float atomics).
  redm[tid] = magsum;
  redc[tid] = cossum;
  __syncthreads();
  for (int s = 128; s > 0; s >>= 1) {
    if (tid < s) { redm[tid] += redm[tid + s]; redc[tid] += redc[tid + s]; }
    __syncthreads();
  }
  if (tid == 0) {
    partials[2 * (int)blockIdx.x + 0] = redm[0];
    partials[2 * (int)blockIdx.x + 1] = redc[0];
  }
}

// Exact fp32 cross-lane sum of 32 per-lane values using V_WMMA_F32_16X16X4_F32:
// A[m][0]=v[lane m], A[m][2]=v[lane 16+m], A[.][1]=A[.][3]=0, B=ones, C=0
// -> D[m][*] = v[m]+v[m+16]; summing the 8 D VGPRs + one shfl_xor(16) gives the total.
__device__ __forceinline__ float wmma_reduce32(float v) {
  v2f a;    a.x = v;    a.y = 0.0f;
  v2f ones; ones.x = 1.0f; ones.y = 1.0f;
  v8f c = {};
  v8f d = __builtin_amdgcn_wmma_f32_16x16x4_f32(false, a, false, ones,
                                                (short)0, c, false, false);
  float s = d[0] + d[1] + d[2] + d[3] + d[4] + d[5] + d[6] + d[7];
  s += __shfl_xor(s, 16, 32);
  return s;
}

__global__ void __launch_bounds__(32)
sobel_loss_finalize(const float* __restrict__ partials, int nparts,
                    float* __restrict__ out, float invN) {
  const int lane = threadIdx.x;  // single wave32, EXEC all ones at the WMMAs
  float m = 0.f, c = 0.f;
  for (int i = lane; i < nparts; i += 32) {
    m += partials[2 * i + 0];
    c += partials[2 * i + 1];
  }
  const float msum = wmma_reduce32(m);
  const float csum = wmma_reduce32(c);
  if (lane == 0) {
    const float mag_loss = msum * invN;
    const float dir_loss = 1.0f - csum * invN;
    out[0] = 0.2f * (mag_loss + dir_loss);
  }
}

extern "C" void kernel_launch(void* const* d_in, const int* in_sizes, int n_in,
                              void* d_out, int out_size, void* d_ws, size_t ws_size,
                              hipStream_t stream) {
  (void)in_sizes; (void)n_in; (void)out_size; (void)ws_size;
  const float* pred = (const float*)d_in[0];
  const float* targ = (const float*)d_in[1];
  float* out = (float*)d_out;
  float* partials = (float*)d_ws;   // NBLOCKS * 2 floats = 12.8 KB

  sobel_loss_tiles<<<NBLOCKS, 256, 0, stream>>>(pred, targ, partials);
  const float invN = 1.0f / (float)((long long)TB * Dd * Hd * Wd);
  sobel_loss_finalize<<<1, 32, 0, stream>>>(partials, NBLOCKS, out, invN);
}